// GraphSAGE_61203283968626
// MI455X (gfx1250) — compile-verified
//
#include <hip/hip_runtime.h>
#include <cstddef>
#include <cstdint>

#define D 128
#define LDS_PP2 144   // float2 pitch per K-pair row: 2*144 = 288 dwords == 32 mod 64 -> conflict-free half-wave split

typedef float v2f __attribute__((ext_vector_type(2)));
typedef float v8f __attribute__((ext_vector_type(8)));

// ---------------------------------------------------------------------------
// f32 WMMA wrapper (16x16x4).  Host pass never executes the stub.
// ---------------------------------------------------------------------------
__device__ __forceinline__ v8f wmma_f32(v2f a, v2f b, v8f c) {
#if defined(__gfx1250__)
  return __builtin_amdgcn_wmma_f32_16x16x4_f32(false, a, false, b,
                                               (short)0, c, false, false);
#else
  c[0] += a[0] * b[0];  // host-compile stub only
  return c;
#endif
}

// ---------------------------------------------------------------------------
// Utility: zero a float buffer
// ---------------------------------------------------------------------------
__global__ void zero_f32(float* __restrict__ p, int n) {
  int i = blockIdx.x * blockDim.x + threadIdx.x;
  if (i < n) p[i] = 0.0f;
}

// ---------------------------------------------------------------------------
// Repack W[n][k] (row-major DxD) into interleaved-transposed layout WTi such
// that, viewed as float2[64][128]:  WTi2[p][n] = { W[n][2p], W[n][2p+1] }.
// Flat mapping: WTi[idx] = W[n*D + k] with n=(idx>>1)&127, k=2*(idx>>8)|(idx&1)
// -> writes perfectly coalesced; GEMM blocks then do a flat float4 LDS fill.
// ---------------------------------------------------------------------------
__global__ void repack_weight(const float* __restrict__ W, float* __restrict__ WTi) {
  int idx = blockIdx.x * blockDim.x + threadIdx.x;
  if (idx >= D * D) return;
  int n = (idx >> 1) & (D - 1);
  int k = ((idx >> 8) << 1) | (idx & 1);
  WTi[idx] = W[n * D + k];
}

// ---------------------------------------------------------------------------
// Edge scatter: one wave32 per edge.  Each lane moves float4 (32*16B = 512B row
// in one coalesced burst) and issues 4 no-return global f32 atomics into the
// mean-aggregation numerator.  Lane 0 bumps the degree (layer 1 only).
// x fits in the 192MB L2 -> gather + atomics are L2-resident.
// ---------------------------------------------------------------------------
__global__ void scatter_edges(const float* __restrict__ x,
                              const int* __restrict__ src,
                              const int* __restrict__ dst,
                              float* __restrict__ msg,
                              float* __restrict__ deg,
                              int nEdges) {
  int wave = (int)((blockIdx.x * blockDim.x + threadIdx.x) >> 5);
  int lane = threadIdx.x & 31;
  if (wave >= nEdges) return;
  int s = src[wave];
  int d = dst[wave];
  float4 v = ((const float4*)(x + (size_t)s * D))[lane];
  float* o = msg + (size_t)d * D + lane * 4;
  atomicAdd(o + 0, v.x);
  atomicAdd(o + 1, v.y);
  atomicAdd(o + 2, v.z);
  atomicAdd(o + 3, v.w);
  if (deg != nullptr && lane == 0) atomicAdd(deg + d, 1.0f);
}

// ---------------------------------------------------------------------------
// One SAGE layer:  out = xin @ Wself^T + (msg/max(deg,1)) @ Wneigh^T + bias
// Block = 128 threads (4 wave32).  Each wave computes a 16-row x 128-col strip
// (8 accumulator tiles of 16x16 f32 = 64 VGPRs) via V_WMMA_F32_16X16X4_F32.
// Two phases (self weights, then neighbor weights) so LDS = 64*144*8 = 73728B
// per block -> 4 blocks/WGP.
// Inner loop is software-pipelined: all 8 B fragments of a k-step are fetched
// (4x ds_load_2addr_b64, one dscnt wait) before the 8 WMMAs issue, and the
// next k-step's A fragment is prefetched under the WMMAs.
// ---------------------------------------------------------------------------
__global__ void __launch_bounds__(128)
sage_layer(const float* __restrict__ xin, const float* __restrict__ msg,
           const float* __restrict__ deg,
           const float* __restrict__ WTiS, const float* __restrict__ WTiN,
           const float* __restrict__ bias,
           float* __restrict__ out, int nNodes) {
  extern __shared__ float lds[];          // float2[64][LDS_PP2] view
  const int tid  = threadIdx.x;
  const int wave = tid >> 5;
  const int lane = tid & 31;
  const int half = lane >> 4;             // which K half of the fragment
  const int l16  = lane & 15;
  const int m0   = blockIdx.x * 64 + wave * 16;
  const int mRow = m0 + l16;              // A-matrix row owned by this lane
  const int mC   = (mRow < nNodes) ? mRow : (nNodes - 1);

  const float invd = 1.0f / fmaxf(deg[mC], 1.0f);
  const float* xr = xin + (size_t)mC * D + half * 2;
  const float* mr = msg + (size_t)mC * D + half * 2;
  const v2f* B2 = (const v2f*)lds;

  v8f acc[8];
#pragma unroll
  for (int t = 0; t < 8; ++t) acc[t] = (v8f)(0.0f);

  // ---- phase 0: self weights ---- phase 1: neighbor weights ----
  for (int phase = 0; phase < 2; ++phase) {
    const float4* src4 = (const float4*)(phase == 0 ? WTiS : WTiN);
    const float*  arow = (phase == 0) ? xr : mr;

    __syncthreads();                       // prior readers done before refill
    for (int i = tid; i < (D * D) / 4; i += 128) {
      // flat float2 f2=2i -> p=f2>>7, n=f2&127 ; dest f2 = p*LDS_PP2 + n
      ((float4*)lds)[(i >> 6) * (LDS_PP2 / 2) + (i & 63)] = src4[i];
    }
    __syncthreads();

    v2f aCur = *(const v2f*)(arow);        // A fragment for k0 = 0
    if (phase == 1) aCur *= invd;

    for (int k0 = 0; k0 < D; k0 += 4) {
      // fetch all 8 B fragments of this k-step first (one dscnt wait total)
      const int base2 = ((k0 >> 1) + half) * LDS_PP2 + l16;
      v2f b[8];
#pragma unroll
      for (int t = 0; t < 8; ++t) b[t] = B2[base2 + t * 16];

      // prefetch next k-step's A fragment under the WMMAs (clamped, no branch)
      const int kN = (k0 + 4 < D) ? (k0 + 4) : k0;
      v2f aNext = *(const v2f*)(arow + kN);
      if (phase == 1) aNext *= invd;

#pragma unroll
      for (int t = 0; t < 8; ++t) acc[t] = wmma_f32(aCur, b[t], acc[t]);
      aCur = aNext;
    }
  }

  // ---- epilogue: bias + store.  C layout: VGPR r holds rows r (lanes 0-15)
  // and r+8 (lanes 16-31); column within tile = lane&15. ----
#pragma unroll
  for (int t = 0; t < 8; ++t) {
    const int col = t * 16 + l16;
    const float bv = bias[col];
#pragma unroll
    for (int r = 0; r < 8; ++r) {
      int row = m0 + r + half * 8;
      if (row < nNodes) out[(size_t)row * D + col] = acc[t][r] + bv;
    }
  }
}

// ---------------------------------------------------------------------------
// Row-wise L2 normalization in place: one wave per node, float4 per lane.
// ---------------------------------------------------------------------------
__global__ void l2_normalize(float* __restrict__ h, int nNodes) {
  int wave = (int)((blockIdx.x * blockDim.x + threadIdx.x) >> 5);
  int lane = threadIdx.x & 31;
  if (wave >= nNodes) return;
  float4* row = (float4*)(h + (size_t)wave * D);
  float4 v = row[lane];
  float ss = v.x * v.x + v.y * v.y + v.z * v.z + v.w * v.w;
#pragma unroll
  for (int off = 16; off > 0; off >>= 1) ss += __shfl_xor(ss, off, 32);
  float scale = 1.0f / fmaxf(sqrtf(ss), 1e-12f);
  v.x *= scale; v.y *= scale; v.z *= scale; v.w *= scale;
  row[lane] = v;
}

// ---------------------------------------------------------------------------
extern "C" void kernel_launch(void* const* d_in, const int* in_sizes, int n_in,
                              void* d_out, int out_size, void* d_ws, size_t ws_size,
                              hipStream_t stream) {
  const float* x   = (const float*)d_in[0];
  const int*   src = (const int*)  d_in[1];
  const int*   dst = (const int*)  d_in[2];
  const float* Ws1 = (const float*)d_in[3];
  const float* Wn1 = (const float*)d_in[4];
  const float* b1  = (const float*)d_in[5];
  const float* Ws2 = (const float*)d_in[6];
  const float* Wn2 = (const float*)d_in[7];
  const float* b2  = (const float*)d_in[8];
  float* out = (float*)d_out;

  const int N = in_sizes[0] / D;      // 100000
  const int E = in_sizes[1];          // 1600000

  // workspace carve-up (~103 MB)
  char* ws = (char*)d_ws;
  float* msg = (float*)ws; ws += (size_t)N * D * sizeof(float);
  float* h1  = (float*)ws; ws += (size_t)N * D * sizeof(float);
  float* deg = (float*)ws; ws += (((size_t)N * sizeof(float)) + 255) & ~(size_t)255;
  float* WTs1 = (float*)ws; ws += (size_t)D * D * sizeof(float);
  float* WTn1 = (float*)ws; ws += (size_t)D * D * sizeof(float);
  float* WTs2 = (float*)ws; ws += (size_t)D * D * sizeof(float);
  float* WTn2 = (float*)ws; ws += (size_t)D * D * sizeof(float);

  // repack all four weight matrices into WMMA-B-friendly interleaved layout
  {
    dim3 blk(256), grd((D * D + 255) / 256);
    repack_weight<<<grd, blk, 0, stream>>>(Ws1, WTs1);
    repack_weight<<<grd, blk, 0, stream>>>(Wn1, WTn1);
    repack_weight<<<grd, blk, 0, stream>>>(Ws2, WTs2);
    repack_weight<<<grd, blk, 0, stream>>>(Wn2, WTn2);
  }

  const size_t ldsBytes = (size_t)64 * LDS_PP2 * 2 * sizeof(float);  // 73728
  dim3 gemmGrid((N + 63) / 64);
  const int nd = N * D;

  // ---- layer 1 ----
  zero_f32<<<(nd + 255) / 256, 256, 0, stream>>>(msg, nd);
  zero_f32<<<(N + 255) / 256, 256, 0, stream>>>(deg, N);
  scatter_edges<<<(E * 32 + 255) / 256, 256, 0, stream>>>(x, src, dst, msg, deg, E);
  sage_layer<<<gemmGrid, 128, ldsBytes, stream>>>(x, msg, deg, WTs1, WTn1, b1, h1, N);

  // ---- layer 2 ----
  zero_f32<<<(nd + 255) / 256, 256, 0, stream>>>(msg, nd);
  scatter_edges<<<(E * 32 + 255) / 256, 256, 0, stream>>>(h1, src, dst, msg, nullptr, E);
  sage_layer<<<gemmGrid, 128, ldsBytes, stream>>>(h1, msg, deg, WTs2, WTn2, b2, out, N);

  // ---- row-wise L2 normalize, in place on d_out ----
  l2_normalize<<<(N * 32 + 255) / 256, 256, 0, stream>>>(out, N);
}